// PhyCell_40699110097119
// MI455X (gfx1250) — compile-verified
//
#include <hip/hip_runtime.h>
#include <math.h>

// ---------------------------------------------------------------------------
// PhyCell for MI455X (gfx1250, wave32).  All heavy contractions go through
// V_WMMA_F32_16X16X4_F32 (fp32 in / fp32 accumulate).  Each wave computes a
// 16x64 C strip (4 accumulators) so the A fragment is reused across 4 WMMAs,
// cutting VMEM instructions per WMMA ~1.6x vs a single-tile wave.
// Working set (~140MB) fits in the 192MB L2.
// Sizes: B=8 T=12 N=512 D=64 KS=3 C_OUT=64 HD=64 S2=6 H=384 GIN=768 3H=1152
// ---------------------------------------------------------------------------

typedef __attribute__((ext_vector_type(2))) float v2f;
typedef __attribute__((ext_vector_type(8))) float v8f;

#define NB   8
#define TT   12
#define NN   512
#define DD   64
#define S2V  6
#define HH   384
#define H3   1152
#define GIN  768
#define KM   1536   // KS*NN

__device__ __forceinline__ float sigf(float x) { return 1.0f / (1.0f + __expf(-x)); }

// ---- wave-level 16x64 strip: C[16x64] += A(16xK,lda) @ B(Kx64 slice, ldb)
__device__ __forceinline__ void wmma_strip_nn(const float* __restrict__ A, int lda,
                                              const float* __restrict__ B, int ldb,
                                              int K, v8f* __restrict__ acc) {
  int l  = threadIdx.x & 31;
  int mn = l & 15;            // A row / B col-within-16 for this lane
  int kk = (l >> 4) << 1;     // 0 or 2
  for (int k0 = 0; k0 < K; k0 += 4) {
    v2f a;
    a[0] = A[mn * lda + k0 + kk];
    a[1] = A[mn * lda + k0 + kk + 1];
#pragma unroll
    for (int j = 0; j < 4; ++j) {
      v2f b;
      b[0] = B[(k0 + kk) * ldb + 16 * j + mn];
      b[1] = B[(k0 + kk + 1) * ldb + 16 * j + mn];
      acc[j] = __builtin_amdgcn_wmma_f32_16x16x4_f32(false, a, false, b, (short)0,
                                                     acc[j], false, false);
    }
  }
}

// ---- wave-level 16x64 strip: C[16x64] += A(16xK,lda) @ Bt(64xK,ldb)^T
__device__ __forceinline__ void wmma_strip_nt(const float* __restrict__ A, int lda,
                                              const float* __restrict__ Bt, int ldb,
                                              int K, v8f* __restrict__ acc) {
  int l  = threadIdx.x & 31;
  int mn = l & 15;
  int kk = (l >> 4) << 1;
  for (int k0 = 0; k0 < K; k0 += 4) {
    v2f a;
    a[0] = A[mn * lda + k0 + kk];
    a[1] = A[mn * lda + k0 + kk + 1];
#pragma unroll
    for (int j = 0; j < 4; ++j) {
      v2f b;
      b[0] = Bt[(16 * j + mn) * ldb + k0 + kk];
      b[1] = Bt[(16 * j + mn) * ldb + k0 + kk + 1];
      acc[j] = __builtin_amdgcn_wmma_f32_16x16x4_f32(false, a, false, b, (short)0,
                                                     acc[j], false, false);
    }
  }
}

__device__ __forceinline__ void wmma_store_strip(float* __restrict__ C, int ldc,
                                                 const v8f* __restrict__ acc) {
  int l  = threadIdx.x & 31;
  int n  = l & 15;
  int mo = (l >> 4) * 8;
#pragma unroll
  for (int j = 0; j < 4; ++j)
    for (int r = 0; r < 8; ++r) C[(mo + r) * ldc + 16 * j + n] = acc[j][r];
}

// ===========================================================================
// 1) Time attention -> attn(b, s2, n)
// ===========================================================================
__global__ void attn_kernel(const float* __restrict__ x1, const float* __restrict__ agg_w,
                            const float* __restrict__ agg_b, const float* __restrict__ time_w,
                            const float* __restrict__ time_b, float* __restrict__ attn) {
  int idx = blockIdx.x * blockDim.x + threadIdx.x;   // b*512+n
  if (idx >= NB * NN) return;
  int b = idx >> 9, n = idx & 511;
  float agg[TT];
  for (int t = 0; t < TT; ++t) {
    const float* p = x1 + (((size_t)b * TT + t) * NN + n) * DD;
    float s = 0.f;
    for (int d = 0; d < DD; ++d) s += p[d] * agg_w[d];
    s += agg_b[0];
    agg[t] = s > 0.f ? s : 0.f;
  }
  float taw[TT], mx = -1e30f;
  for (int t = 0; t < TT; ++t) {
    float s = time_b[t];
    for (int u = 0; u < TT; ++u) s += agg[u] * time_w[t * TT + u];
    taw[t] = s;
    mx = fmaxf(mx, s);
  }
  float den = 0.f;
  for (int t = 0; t < TT; ++t) { taw[t] = __expf(taw[t] - mx); den += taw[t]; }
  float inv = 1.f / den;
  for (int s = 0; s < S2V; ++s)
    attn[((size_t)b * S2V + s) * NN + n] = (taw[2 * s] + taw[2 * s + 1]) * inv;
}

// ===========================================================================
// 2) g(b,i,j) = sum_s in2(b,s,i,j)*attn(b,s,i)
// ===========================================================================
__global__ void gbuild_kernel(const float* __restrict__ in2, const float* __restrict__ attn,
                              float* __restrict__ g) {
  int j = threadIdx.x, i = blockIdx.x, b = blockIdx.y;
  float acc = 0.f;
  for (int s = 0; s < S2V; ++s)
    acc += in2[(((size_t)b * S2V + s) * NN + i) * NN + j] *
           attn[((size_t)b * S2V + s) * NN + i];
  g[(((size_t)b * NN) + i) * NN + j] = acc;
}

// 3) column L1 norms
__global__ void colnorm_kernel(const float* __restrict__ g, float* __restrict__ nrm) {
  int j = threadIdx.x, b = blockIdx.x;
  const float* G = g + (size_t)b * NN * NN;
  float acc = 0.f;
  for (int i = 0; i < NN; ++i) acc += fabsf(G[i * NN + j]);
  nrm[b * NN + j] = fmaxf(acc, 1e-12f);
}

// 4) g <- g/nrm + I + graph_all (in place), dg(b,i) = row sums
__global__ void gfin_dg_kernel(float* __restrict__ g, const float* __restrict__ nrm,
                               const float* __restrict__ graph_all, float* __restrict__ dg) {
  __shared__ float red[NN];
  int j = threadIdx.x, i = blockIdx.x, b = blockIdx.y;
  size_t idx = (((size_t)b * NN) + i) * NN + j;
  float v = g[idx] / nrm[b * NN + j] + (i == j ? 1.f : 0.f) + graph_all[i * NN + j];
  g[idx] = v;
  red[j] = v;
  __syncthreads();
  for (int st = 256; st > 0; st >>= 1) {
    if (j < st) red[j] += red[j + st];
    __syncthreads();
  }
  if (j == 0) dg[b * NN + i] = red[0];
}

// 5) L(i,j) = ((i==j)? dg_i : -g_ij) / sqrt(dg_i*dg_j)
__global__ void lap_kernel(const float* __restrict__ g, const float* __restrict__ dg,
                           float* __restrict__ L) {
  int j = threadIdx.x, i = blockIdx.x, b = blockIdx.y;
  size_t idx = (((size_t)b * NN) + i) * NN + j;
  float dgi = dg[b * NN + i], dgj = dg[b * NN + j];
  float v = (i == j) ? dgi : -g[idx];
  L[idx] = v * rsqrtf(dgi * dgj);
}

// 6) dominant eigenvalue by power iteration + Rayleigh quotient (per batch)
__global__ void eig_kernel(const float* __restrict__ L, float* __restrict__ lam) {
  __shared__ float x[NN];
  __shared__ float red[NN];
  int b = blockIdx.x, i = threadIdx.x;
  const float* Lb = L + (size_t)b * NN * NN;
  x[i] = 0.044194173f;   // 1/sqrt(512)
  __syncthreads();
  for (int it = 0; it < 64; ++it) {
    const float* row = Lb + (size_t)i * NN;
    float y = 0.f;
    for (int j = 0; j < NN; ++j) y += row[j] * x[j];
    __syncthreads();                 // everyone done reading x
    red[i] = y * y;
    __syncthreads();
    for (int st = 256; st > 0; st >>= 1) {
      if (i < st) red[i] += red[i + st];
      __syncthreads();
    }
    float rinv = rsqrtf(red[0] + 1e-30f);
    __syncthreads();
    x[i] = y * rinv;
    __syncthreads();
  }
  // Rayleigh quotient (||x||=1): lam = x . (L x)
  const float* row = Lb + (size_t)i * NN;
  float y = 0.f;
  for (int j = 0; j < NN; ++j) y += row[j] * x[j];
  __syncthreads();
  red[i] = x[i] * y;
  __syncthreads();
  for (int st = 256; st > 0; st >>= 1) {
    if (i < st) red[i] += red[i + st];
    __syncthreads();
  }
  if (i == 0) lam[b] = red[0];
}

// 7) LkA[b][n][k*512+m]: k=0 -> I, k=1 -> 2L/lam - I  (k=2 filled by cheb2)
__global__ void lkfill_kernel(const float* __restrict__ L, const float* __restrict__ lam,
                              float* __restrict__ LkA) {
  int m = threadIdx.x, n = blockIdx.x, b = blockIdx.y;
  float lh = 2.f * L[(((size_t)b * NN) + n) * NN + m] / lam[b] - (n == m ? 1.f : 0.f);
  size_t base = (((size_t)b * NN) + n) * KM;
  LkA[base + m] = (n == m) ? 1.f : 0.f;
  LkA[base + NN + m] = lh;
}

// 8) Cheb T2 = 2*Lh@Lh - I  (WMMA, one wave per 16x64 strip)
__global__ void cheb2_kernel(float* __restrict__ LkA) {
  int b = blockIdx.z;
  const float* Lh = LkA + (size_t)b * NN * KM + NN;       // k=1 slice, ld=KM
  int tm = blockIdx.y * 16, tn = blockIdx.x * 64;
  v8f acc[4] = {};
  wmma_strip_nn(Lh + (size_t)tm * KM, KM, Lh + tn, KM, NN, acc);
  float* C = LkA + (size_t)b * NN * KM + 2 * NN + (size_t)tm * KM + tn;  // k=2 slice
  int l = threadIdx.x & 31, n = l & 15, mo = (l >> 4) * 8;
#pragma unroll
  for (int j = 0; j < 4; ++j)
    for (int r = 0; r < 8; ++r) {
      int gm = tm + mo + r, gn = tn + 16 * j + n;
      C[(mo + r) * KM + 16 * j + n] = 2.f * acc[j][r] - (gm == gn ? 1.f : 0.f);
    }
}

// 9) theta repack: thetaR[k][i][o] = theta[i,o,k]
__global__ void thetaR_kernel(const float* __restrict__ theta, float* __restrict__ thetaR) {
  int idx = blockIdx.x * blockDim.x + threadIdx.x;  // k*4096 + i*64 + o
  if (idx >= 3 * DD * DD) return;
  int k = idx >> 12, i = (idx >> 6) & 63, o = idx & 63;
  thetaR[idx] = theta[i * 192 + o * 3 + k];
}

// 10) stage-A GEMM: y2[(b,t,k)][m][o] = x1[b,t] (512x64) @ thetaR[k] (64x64)
__global__ void gemmA_kernel(const float* __restrict__ x1, const float* __restrict__ thetaR,
                             float* __restrict__ y2) {
  int z = blockIdx.z;                       // (b*12+t)*3 + k
  const float* A = x1 + (size_t)(z / 3) * (NN * DD);
  const float* B = thetaR + (size_t)(z % 3) * (DD * DD);
  float* C = y2 + (size_t)z * (NN * DD);
  int tm = blockIdx.y * 16;
  v8f acc[4] = {};
  wmma_strip_nn(A + tm * DD, DD, B, DD, DD, acc);
  wmma_store_strip(C + (size_t)tm * DD, DD, acc);
}

// 11) stage-B GEMM + bias + relu -> gx[(b*512+n)][t*64+o]
__global__ void gemmB_kernel(const float* __restrict__ LkA, const float* __restrict__ y2,
                             const float* __restrict__ bgc, float* __restrict__ gx) {
  int z = blockIdx.z;                       // b*12 + t
  int b = z / TT, t = z % TT;
  const float* A = LkA + (size_t)b * NN * KM;
  const float* B = y2 + (size_t)z * (KM * DD);
  int tm = blockIdx.y * 16;
  v8f acc[4] = {};
  wmma_strip_nn(A + (size_t)tm * KM, KM, B, DD, KM, acc);
  float* C = gx + ((size_t)b * NN + tm) * GIN + t * DD;
  int l = threadIdx.x & 31, n = l & 15, mo = (l >> 4) * 8;
#pragma unroll
  for (int j = 0; j < 4; ++j)
    for (int r = 0; r < 8; ++r) {
      float v = acc[j][r] + bgc[16 * j + n];
      C[(mo + r) * GIN + 16 * j + n] = v > 0.f ? v : 0.f;
    }
}

// 12) hin[(b*512+n)][h] = h_input[b][h][n]
__global__ void hin_kernel(const float* __restrict__ hin_src, float* __restrict__ hin) {
  int idx = blockIdx.x * blockDim.x + threadIdx.x;
  if (idx >= NB * NN * HH) return;
  int r = idx / HH, h = idx % HH;
  int b = r >> 9, n = r & 511;
  hin[idx] = hin_src[((size_t)b * HH + h) * NN + n];
}

// 13) generic NT GEMM: C(MxN) = A(MxK) @ W(NxK)^T  (biases added later)
__global__ void gemm_nt_kernel(const float* __restrict__ A, int lda,
                               const float* __restrict__ W, int ldw,
                               float* __restrict__ C, int ldc, int K) {
  int tm = blockIdx.y * 16, tn = blockIdx.x * 64;
  v8f acc[4] = {};
  wmma_strip_nt(A + (size_t)tm * lda, lda, W + (size_t)tn * ldw, ldw, K, acc);
  wmma_store_strip(C + (size_t)tm * ldc + tn, ldc, acc);
}

// 14) GRU gates -> h1[b][s][n][hd] (first chunk of d_out)
__global__ void gate_kernel(const float* __restrict__ gi, const float* __restrict__ gh,
                            const float* __restrict__ b_ih, const float* __restrict__ b_hh,
                            const float* __restrict__ hin, float* __restrict__ h1) {
  int idx = blockIdx.x * blockDim.x + threadIdx.x;   // r*384 + c
  if (idx >= NB * NN * HH) return;
  int r = idx / HH, c = idx % HH;
  size_t gr = (size_t)r * H3;
  float ir = gi[gr + c] + b_ih[c];
  float iz = gi[gr + HH + c] + b_ih[HH + c];
  float ig = gi[gr + 2 * HH + c] + b_ih[2 * HH + c];
  float hr = gh[gr + c] + b_hh[c];
  float hz = gh[gr + HH + c] + b_hh[HH + c];
  float hg = gh[gr + 2 * HH + c] + b_hh[2 * HH + c];
  float rg = sigf(ir + hr);
  float zg = sigf(iz + hz);
  float cg = tanhf(ig + rg * hg);
  float h = (1.f - zg) * cg + zg * hin[(size_t)r * HH + c];
  int b = r >> 9, n = r & 511;
  int s = c >> 6, hd = c & 63;
  h1[(((size_t)b * S2V + s) * NN + n) * DD + hd] = h;
}

// 15) arr: sigmoid(h1 . arr_w + arr_b) per (b,s,n)
__global__ void arr_kernel(const float* __restrict__ h1, const float* __restrict__ arr_w,
                           const float* __restrict__ arr_b, float* __restrict__ arrA,
                           float* __restrict__ arrB) {
  int idx = blockIdx.x * blockDim.x + threadIdx.x;  // b*6*512
  if (idx >= NB * S2V * NN) return;
  const float* h = h1 + (size_t)idx * DD;
  float s0 = 0.f, s1 = 0.f;
  for (int d = 0; d < DD; ++d) {
    s0 += h[d] * arr_w[d];
    s1 += h[d] * arr_w[DD + d];
  }
  arrA[idx] = sigf(s0 + arr_b[0]);
  arrB[idx] = sigf(s1 + arr_b[1]);
}

// 16) u(b,s,j) = sum_m in2(b,s,m,j)
__global__ void usum_kernel(const float* __restrict__ in2, float* __restrict__ u) {
  int bs = blockIdx.x, j = threadIdx.x;
  const float* A = in2 + (size_t)bs * NN * NN;
  float acc = 0.f;
  for (int m = 0; m < NN; ++m) acc += A[(size_t)m * NN + j];
  u[(size_t)bs * NN + j] = acc;
}

// 17) phy(b,n) = sum_s (u(b,s).arrA(b,s)) * arrB(b,s,n)
__global__ void phy_kernel(const float* __restrict__ u, const float* __restrict__ arrA,
                           const float* __restrict__ arrB, float* __restrict__ phy) {
  __shared__ float red[NN];
  __shared__ float sd[S2V];
  int b = blockIdx.x, n = threadIdx.x;
  for (int s = 0; s < S2V; ++s) {
    size_t o = ((size_t)b * S2V + s) * NN + n;
    red[n] = u[o] * arrA[o];
    __syncthreads();
    for (int st = 256; st > 0; st >>= 1) {
      if (n < st) red[n] += red[n + st];
      __syncthreads();
    }
    if (n == 0) sd[s] = red[0];
    __syncthreads();
  }
  float acc = 0.f;
  for (int s = 0; s < S2V; ++s) acc += sd[s] * arrB[((size_t)b * S2V + s) * NN + n];
  phy[b * NN + n] = acc;
}

// ===========================================================================
extern "C" void kernel_launch(void* const* d_in, const int* in_sizes, int n_in,
                              void* d_out, int out_size, void* d_ws, size_t ws_size,
                              hipStream_t stream) {
  (void)in_sizes; (void)n_in; (void)out_size; (void)ws_size;
  const float* inputs_1  = (const float*)d_in[0];
  const float* inputs_2  = (const float*)d_in[1];
  const float* h_input   = (const float*)d_in[2];
  const float* graph_all = (const float*)d_in[3];
  const float* agg_w     = (const float*)d_in[4];
  const float* agg_b     = (const float*)d_in[5];
  const float* time_w    = (const float*)d_in[6];
  const float* time_b    = (const float*)d_in[7];
  const float* theta     = (const float*)d_in[8];
  const float* b_gc      = (const float*)d_in[9];
  const float* w_ih      = (const float*)d_in[10];
  const float* w_hh      = (const float*)d_in[11];
  const float* b_ih      = (const float*)d_in[12];
  const float* b_hh      = (const float*)d_in[13];
  const float* arr_w     = (const float*)d_in[14];
  const float* arr_b     = (const float*)d_in[15];

  float* h1  = (float*)d_out;                      // (8,6,512,64)
  float* phy = h1 + (size_t)NB * S2V * NN * DD;    // (8,512)

  // workspace layout (floats); total ~34.2M floats (~137 MB)
  float* ws = (float*)d_ws;
  size_t o = 0;
  float* attn   = ws + o; o += (size_t)NB * S2V * NN;        // 24576
  float* g      = ws + o; o += (size_t)NB * NN * NN;         // 2097152
  float* nrm    = ws + o; o += (size_t)NB * NN;              // 4096
  float* dg     = ws + o; o += (size_t)NB * NN;              // 4096
  float* L      = ws + o; o += (size_t)NB * NN * NN;         // 2097152
  float* lam    = ws + o; o += 32;                           // 8 (+pad)
  float* LkA    = ws + o; o += (size_t)NB * NN * KM;         // 6291456
  float* thetaR = ws + o; o += (size_t)3 * DD * DD;          // 12288
  float* y2     = ws + o; o += (size_t)NB * TT * 3 * NN * DD;// 9437184
  float* gx     = ws + o; o += (size_t)NB * NN * GIN;        // 3145728
  float* hin    = ws + o; o += (size_t)NB * NN * HH;         // 1572864
  float* gi     = ws + o; o += (size_t)NB * NN * H3;         // 4718592
  float* gh     = ws + o; o += (size_t)NB * NN * H3;         // 4718592
  float* arrA   = ws + o; o += (size_t)NB * S2V * NN;        // 24576
  float* arrB   = ws + o; o += (size_t)NB * S2V * NN;        // 24576
  float* u      = ws + o; o += (size_t)NB * S2V * NN;        // 24576

  // ---- graph / Laplacian path ----
  attn_kernel<<<16, 256, 0, stream>>>(inputs_1, agg_w, agg_b, time_w, time_b, attn);
  gbuild_kernel<<<dim3(NN, NB), NN, 0, stream>>>(inputs_2, attn, g);
  colnorm_kernel<<<NB, NN, 0, stream>>>(g, nrm);
  gfin_dg_kernel<<<dim3(NN, NB), NN, 0, stream>>>(g, nrm, graph_all, dg);
  lap_kernel<<<dim3(NN, NB), NN, 0, stream>>>(g, dg, L);
  eig_kernel<<<NB, NN, 0, stream>>>(L, lam);
  lkfill_kernel<<<dim3(NN, NB), NN, 0, stream>>>(L, lam, LkA);
  cheb2_kernel<<<dim3(8, 32, NB), 32, 0, stream>>>(LkA);

  // ---- graph convolution as two WMMA GEMM stages ----
  thetaR_kernel<<<48, 256, 0, stream>>>(theta, thetaR);
  gemmA_kernel<<<dim3(1, 32, NB * TT * 3), 32, 0, stream>>>(inputs_1, thetaR, y2);
  gemmB_kernel<<<dim3(1, 32, NB * TT), 32, 0, stream>>>(LkA, y2, b_gc, gx);

  // ---- GRU ----
  hin_kernel<<<6144, 256, 0, stream>>>(h_input, hin);
  gemm_nt_kernel<<<dim3(H3 / 64, (NB * NN) / 16), 32, 0, stream>>>(gx, GIN, w_ih, GIN,
                                                                   gi, H3, GIN);
  gemm_nt_kernel<<<dim3(H3 / 64, (NB * NN) / 16), 32, 0, stream>>>(hin, HH, w_hh, HH,
                                                                   gh, H3, HH);
  gate_kernel<<<6144, 256, 0, stream>>>(gi, gh, b_ih, b_hh, hin, h1);

  // ---- decay / physical aggregation (rank-1 simplification) ----
  arr_kernel<<<96, 256, 0, stream>>>(h1, arr_w, arr_b, arrA, arrB);
  usum_kernel<<<NB * S2V, NN, 0, stream>>>(inputs_2, u);
  phy_kernel<<<NB, NN, 0, stream>>>(u, arrA, arrB, phy);
}